// routerattn_38817914421575
// MI455X (gfx1250) — compile-verified
//
#include <hip/hip_runtime.h>
#include <hip/hip_bf16.h>

typedef __attribute__((ext_vector_type(2))) float v2f;
typedef __attribute__((ext_vector_type(8))) float v8f;

#define B_   32
#define N_   512
#define D_   1024
#define H_   16
#define HD_  64
#define SEG_ 64

// ---------------- CDNA5 async global->LDS support (guarded) ----------------
#if defined(__HIP_DEVICE_COMPILE__) && \
    __has_builtin(__builtin_amdgcn_global_load_async_to_lds_b128)
#define HAVE_ASYNC 1
#else
#define HAVE_ASYNC 0
#endif

#if HAVE_ASYNC
typedef int v4i __attribute__((vector_size(16)));
typedef __attribute__((address_space(1))) void GVoid;
typedef __attribute__((address_space(3))) void LVoid;
typedef __attribute__((address_space(1))) v4i GV4i;
typedef __attribute__((address_space(3))) v4i LV4i;

__device__ __forceinline__ void async_cp16(const float* g, float* l) {
  // per-lane 16B copy global -> LDS, tracked by ASYNCcnt (no VGPR staging)
  GVoid* gv = (GVoid*)g;   // addrspacecast generic -> global
  LVoid* lv = (LVoid*)l;   // addrspacecast generic -> LDS
  __builtin_amdgcn_global_load_async_to_lds_b128((GV4i*)gv, (LV4i*)lv, 0, 0);
}
#if __has_builtin(__builtin_amdgcn_s_wait_asynccnt)
#define WAIT_ASYNC(n) __builtin_amdgcn_s_wait_asynccnt(n)
#else
#define WAIT_ASYNC(n) asm volatile("s_wait_asynccnt " #n ::: "memory")
#endif
#endif

// ---------------------------------------------------------------------------
// GEMM: C(MxN) = A(MxK) @ W(KxN) + bias, fp32 via V_WMMA_F32_16X16X4_F32
// Block tile 64x128, K-tile 16, 256 threads (8 waves), each wave -> 32x32.
// Double-buffered async global->LDS tile fetch when available.
// ---------------------------------------------------------------------------
#define TM 64
#define TN 128
#define TK 16

__global__ __launch_bounds__(256) void gemm_bias_wmma(
    const float* __restrict__ A, const float* __restrict__ W,
    const float* __restrict__ bias, float* __restrict__ C,
    int M, int N, int K)
{
  // rows 16B-aligned (96B / 528B strides) so async B128 LDS writes are legal
  __shared__ __align__(16) float As[2][TM][TK + 8];   // (m,k)
  __shared__ __align__(16) float Bs[2][TK][TN + 4];   // (k,n) natural order

  const int tid  = threadIdx.x;
  const int wave = tid >> 5;
  const int lane = tid & 31;
  const int hf   = lane >> 4;       // wave half: selects K pair / M+8
  const int li   = lane & 15;
  const int wm   = wave >> 2;       // 0..1  wave row
  const int wn   = wave & 3;        // 0..3  wave col

  const int bM = blockIdx.y * TM;
  const int bN = blockIdx.x * TN;

  // cooperative tile-fetch assignments (16B chunks)
  const int ar = tid >> 2;          // A row 0..63
  const int ac = (tid & 3) * 4;     // A col 0/4/8/12
  const int bk = tid >> 4;          // B k-row 0..15
  const int bn = (tid & 15) * 8;    // B col 0..120 (two 16B chunks)

  v8f acc[2][2] = {};

  auto compute_tile = [&](int buf) {
#pragma unroll
    for (int km = 0; km < 4; ++km) {
      const int k0 = km * 4 + hf * 2;
      v2f a0 = *(const v2f*)&As[buf][wm * 32 +      li][k0];
      v2f a1 = *(const v2f*)&As[buf][wm * 32 + 16 + li][k0];
      v2f b0, b1;
      b0[0] = Bs[buf][k0    ][wn * 32 +      li];
      b0[1] = Bs[buf][k0 + 1][wn * 32 +      li];
      b1[0] = Bs[buf][k0    ][wn * 32 + 16 + li];
      b1[1] = Bs[buf][k0 + 1][wn * 32 + 16 + li];
      acc[0][0] = __builtin_amdgcn_wmma_f32_16x16x4_f32(false, a0, false, b0, (short)0, acc[0][0], false, false);
      acc[0][1] = __builtin_amdgcn_wmma_f32_16x16x4_f32(false, a0, false, b1, (short)0, acc[0][1], false, false);
      acc[1][0] = __builtin_amdgcn_wmma_f32_16x16x4_f32(false, a1, false, b0, (short)0, acc[1][0], false, false);
      acc[1][1] = __builtin_amdgcn_wmma_f32_16x16x4_f32(false, a1, false, b1, (short)0, acc[1][1], false, false);
    }
  };

#if HAVE_ASYNC
  auto issue_tile = [&](int buf, int kb) {   // 3 async ops per wave
    async_cp16(A + (size_t)(bM + ar) * K + (kb + ac),     &As[buf][ar][ac]);
    async_cp16(W + (size_t)(kb + bk) * N + (bN + bn),     &Bs[buf][bk][bn]);
    async_cp16(W + (size_t)(kb + bk) * N + (bN + bn + 4), &Bs[buf][bk][bn + 4]);
  };

  issue_tile(0, 0);
  int buf = 0;
  for (int kb = 0; kb < K; kb += TK) {
    if (kb + TK < K) {
      issue_tile(buf ^ 1, kb + TK);   // prefetch next K-tile into other buffer
      WAIT_ASYNC(3);                  // oldest 3 (current buffer) complete
    } else {
      WAIT_ASYNC(0);
    }
    __syncthreads();                  // all waves' portions visible
    compute_tile(buf);
    __syncthreads();                  // done reading before next overwrite
    buf ^= 1;
  }
#else
  for (int kb = 0; kb < K; kb += TK) {
    float4 va = *(const float4*)(A + (size_t)(bM + ar) * K + (kb + ac));
    float4 v0 = *(const float4*)(W + (size_t)(kb + bk) * N + (bN + bn));
    float4 v1 = *(const float4*)(W + (size_t)(kb + bk) * N + (bN + bn + 4));
    if (kb + TK < K) {
      __builtin_prefetch(A + (size_t)(bM + ar) * K + (kb + TK + ac), 0, 0);
      __builtin_prefetch(W + (size_t)(kb + TK + bk) * N + (bN + bn), 0, 0);
    }
    *(float4*)&As[0][ar][ac]     = va;
    *(float4*)&Bs[0][bk][bn]     = v0;
    *(float4*)&Bs[0][bk][bn + 4] = v1;
    __syncthreads();
    compute_tile(0);
    __syncthreads();
  }
#endif

  // epilogue: lane holds C[m = i + 8*hf][n = li] per 16x16 tile
#pragma unroll
  for (int r = 0; r < 2; ++r) {
#pragma unroll
    for (int c = 0; c < 2; ++c) {
      const int col = bN + wn * 32 + c * 16 + li;
      const float bv = bias[col];
#pragma unroll
      for (int i = 0; i < 8; ++i) {
        const int row = bM + wm * 32 + r * 16 + i + 8 * hf;
        C[(size_t)row * N + col] = acc[r][c][i] + bv;
      }
    }
  }
}

// ---------------------------------------------------------------------------
// Router attention: per (b,n):
//   scores(16x64) = Q_n(16x64) @ R_n(64x64)^T ; softmax over SEG ;
//   received(16x64) = attn @ R_n ; write attn + received (in-place over Q ws).
// One wave per (b,n); 2 waves per block; WMMA f32 16x16x4 both passes.
// ---------------------------------------------------------------------------
__global__ __launch_bounds__(64) void router_attn_wmma(
    const float* __restrict__ Qin,   // (B*N, D) = q projection
    const float* __restrict__ Rt,    // (SEG, N, HD) router
    float* __restrict__ attn_out,    // (B, H, N, SEG)
    float* __restrict__ recv_out)    // (B*N, D), may alias Qin
{
  __shared__ __align__(16) float Rs [2][SEG_][HD_ + 4];   // 272B rows, 16B aligned
  __shared__ __align__(16) float Qs [2][H_  ][HD_ + 4];
  __shared__ __align__(16) float Ssm[2][H_  ][SEG_ + 2];

  const int w    = threadIdx.x >> 5;
  const int lane = threadIdx.x & 31;
  const int hf   = lane >> 4;
  const int li   = lane & 15;

  const int row = blockIdx.x * 2 + w;     // 0 .. B*N-1
  const int b   = row / N_;
  const int n   = row % N_;

  const float* qrow = Qin + (size_t)row * D_;

#if HAVE_ASYNC
  // stage Q row (16x64) and R[:, n, :] (64x64) via async copies (per-wave)
#pragma unroll
  for (int it = 0; it < 8; ++it) {
    const int chunk = it * 32 + lane;             // 0..255
    const int h = chunk >> 4, d4 = (chunk & 15) * 4;
    async_cp16(qrow + h * HD_ + d4, &Qs[w][h][d4]);
  }
#pragma unroll 4
  for (int it = 0; it < 32; ++it) {
    const int chunk = it * 32 + lane;             // 0..1023
    const int s = chunk >> 4, d4 = (chunk & 15) * 4;
    async_cp16(Rt + (size_t)s * (N_ * HD_) + (size_t)n * HD_ + d4, &Rs[w][s][d4]);
  }
  WAIT_ASYNC(0);
  __syncthreads();
#else
#pragma unroll
  for (int it = 0; it < 8; ++it) {
    const int e = (it * 32 + lane) * 4;
    float4 v = *(const float4*)(qrow + e);
    const int h = e >> 6, d = e & 63;
    *(float4*)&Qs[w][h][d] = v;
  }
#pragma unroll 4
  for (int it = 0; it < 32; ++it) {
    const int e = (it * 32 + lane) * 4;
    const int s = e >> 6, d = e & 63;
    float4 v = *(const float4*)(Rt + (size_t)s * (N_ * HD_) + (size_t)n * HD_ + d);
    *(float4*)&Rs[w][s][d] = v;
  }
  __syncthreads();
#endif

  // pass 1: scores = Q @ R^T  (4 tiles over SEG, K = HD = 64)
  v8f sc[4] = {};
#pragma unroll
  for (int km = 0; km < 16; ++km) {
    const int k0 = km * 4 + hf * 2;       // d pair
    v2f a = *(const v2f*)&Qs[w][li][k0];
#pragma unroll
    for (int t = 0; t < 4; ++t) {
      v2f bfrag = *(const v2f*)&Rs[w][t * 16 + li][k0];
      sc[t] = __builtin_amdgcn_wmma_f32_16x16x4_f32(false, a, false, bfrag, (short)0, sc[t], false, false);
    }
  }

  // softmax over SEG (per head row); attn -> global + LDS
#pragma unroll
  for (int i = 0; i < 8; ++i) {
    const int h = i + 8 * hf;
    float mx = fmaxf(fmaxf(sc[0][i], sc[1][i]), fmaxf(sc[2][i], sc[3][i]));
#pragma unroll
    for (int off = 1; off < 16; off <<= 1)
      mx = fmaxf(mx, __shfl_xor(mx, off, 32));
    float ex0 = __expf(sc[0][i] - mx);
    float ex1 = __expf(sc[1][i] - mx);
    float ex2 = __expf(sc[2][i] - mx);
    float ex3 = __expf(sc[3][i] - mx);
    float sum = ex0 + ex1 + ex2 + ex3;
#pragma unroll
    for (int off = 1; off < 16; off <<= 1)
      sum += __shfl_xor(sum, off, 32);
    const float inv = 1.0f / sum;
    const float a0 = ex0 * inv, a1 = ex1 * inv, a2 = ex2 * inv, a3 = ex3 * inv;
    const size_t abase = (((size_t)b * H_ + h) * N_ + n) * SEG_;
    attn_out[abase +  0 + li] = a0;  Ssm[w][h][ 0 + li] = a0;
    attn_out[abase + 16 + li] = a1;  Ssm[w][h][16 + li] = a1;
    attn_out[abase + 32 + li] = a2;  Ssm[w][h][32 + li] = a2;
    attn_out[abase + 48 + li] = a3;  Ssm[w][h][48 + li] = a3;
  }

  // pass 2: received = attn(16x64) @ R(64x64)  (4 tiles over HD, K = SEG)
  v8f rc[4] = {};
#pragma unroll
  for (int km = 0; km < 16; ++km) {
    const int k0 = km * 4 + hf * 2;       // s pair
    v2f a = *(const v2f*)&Ssm[w][li][k0];
#pragma unroll
    for (int t = 0; t < 4; ++t) {
      v2f bfrag;
      bfrag[0] = Rs[w][k0    ][t * 16 + li];
      bfrag[1] = Rs[w][k0 + 1][t * 16 + li];
      rc[t] = __builtin_amdgcn_wmma_f32_16x16x4_f32(false, a, false, bfrag, (short)0, rc[t], false, false);
    }
  }

  float* rrow = recv_out + (size_t)row * D_;
#pragma unroll
  for (int t = 0; t < 4; ++t) {
#pragma unroll
    for (int i = 0; i < 8; ++i) {
      const int h = i + 8 * hf;
      rrow[h * HD_ + t * 16 + li] = rc[t][i];
    }
  }
}

// ---------------------------------------------------------------------------
extern "C" void kernel_launch(void* const* d_in, const int* in_sizes, int n_in,
                              void* d_out, int out_size, void* d_ws, size_t ws_size,
                              hipStream_t stream) {
  (void)in_sizes; (void)n_in; (void)out_size; (void)ws_size;

  const float* query  = (const float*)d_in[0];
  const float* router = (const float*)d_in[3];
  const float* Wq     = (const float*)d_in[4];
  const float* bq     = (const float*)d_in[5];
  const float* Wo     = (const float*)d_in[10];
  const float* bo     = (const float*)d_in[11];
  // k/v projections are dead code w.r.t. the reference outputs -> skipped.

  float* out  = (float*)d_out;                        // (B, N, D)
  float* attn = out + (size_t)B_ * N_ * D_;           // (B, H, N, SEG)
  float* ws   = (float*)d_ws;                         // (B*N, D) scratch

  const int M = B_ * N_;   // 16384
  const int N = D_;        // 1024
  const int K = D_;        // 1024

  dim3 ggrid(N / TN, M / TM);   // (8, 256)

  // q = query @ Wq + bq  -> ws
  gemm_bias_wmma<<<ggrid, 256, 0, stream>>>(query, Wq, bq, ws, M, N, K);
  // router attention: attn -> d_out tail, received -> ws (in place)
  router_attn_wmma<<<(B_ * N_) / 2, 64, 0, stream>>>(ws, router, attn, ws);
  // output = received @ Wo + bo -> d_out head
  gemm_bias_wmma<<<ggrid, 256, 0, stream>>>(ws, Wo, bo, out, M, N, K);
}